// HistogramLayer_28063316312104
// MI455X (gfx1250) — compile-verified
//
#include <hip/hip_runtime.h>

typedef __attribute__((ext_vector_type(2))) float v2f;
typedef __attribute__((ext_vector_type(8))) float v8f;

#define B_   32
#define C_   512
#define L_   4096
#define NB   16
#define EPSV 1e-8f

// workspace layout (bytes)
#define WS_G    0                         // B_*C_ floats (64 KB)
#define WS_GN   (B_ * C_ * 4)             // B_ floats
#define WS_KEYS (WS_GN + 128)             // 2 uints (min/max keys)
#define WS_COS  (WS_KEYS + 128)           // B_*L_ floats (512 KB)

__device__ __forceinline__ unsigned enc_key(float f) {
    unsigned u = __float_as_uint(f);
    return (u & 0x80000000u) ? ~u : (u | 0x80000000u);
}
__device__ __forceinline__ float dec_key(unsigned k) {
    unsigned u = (k & 0x80000000u) ? (k & 0x7FFFFFFFu) : ~k;
    return __uint_as_float(u);
}

// ---- K0: reset min/max accumulators (every call; graph-safe) ----
__global__ void k_init(unsigned* __restrict__ keys) {
    if (threadIdx.x == 0) { keys[0] = 0xFFFFFFFFu; keys[1] = 0u; }
}

// ---- K1: g[b,c] = mean over L. One wave per row, float4 streaming loads ----
__global__ void k_mean(const float* __restrict__ x, float* __restrict__ g) {
    int wave = threadIdx.x >> 5, lane = threadIdx.x & 31;
    int row  = blockIdx.x * 8 + wave;                 // row in [0, B_*C_)
    const float4* xr = (const float4*)(x + (size_t)row * L_);
    float s = 0.f;
    for (int i = lane; i < L_ / 4; i += 32) {
        float4 v = xr[i];
        s += (v.x + v.y) + (v.z + v.w);
    }
    #pragma unroll
    for (int off = 16; off > 0; off >>= 1) s += __shfl_xor(s, off, 32);
    if (lane == 0) g[row] = s * (1.0f / (float)L_);
}

// ---- K2: gn[b] = sqrt(sum_c g^2) ----
__global__ void k_gn(const float* __restrict__ g, float* __restrict__ gn) {
    int b = blockIdx.x, t = threadIdx.x;
    float a  = g[b * C_ + t];
    float c2 = g[b * C_ + 256 + t];
    float s  = a * a + c2 * c2;
    #pragma unroll
    for (int off = 16; off > 0; off >>= 1) s += __shfl_xor(s, off, 32);
    __shared__ float red[8];
    if ((t & 31) == 0) red[t >> 5] = s;
    __syncthreads();
    if (t == 0) {
        float tot = 0.f;
        #pragma unroll
        for (int i = 0; i < 8; ++i) tot += red[i];
        gn[b] = sqrtf(tot);
    }
}

// ---- K3: cos[b,l] via V_WMMA_F32_16X16X4_F32 over the channel dim +
//          fused global min/max (ordered-uint atomics) ----
// grid = (L_/128, B_), block = 256 (8 waves); wave handles a 16-column L tile.
__global__ void k_cos(const float* __restrict__ x, const float* __restrict__ g,
                      const float* __restrict__ gn, float* __restrict__ cosb,
                      unsigned* __restrict__ keys) {
    __shared__ float sg[C_];
    int b = blockIdx.y;
    for (int i = threadIdx.x; i < C_; i += 256) sg[i] = g[b * C_ + i];
    __syncthreads();

    int wave = threadIdx.x >> 5, lane = threadIdx.x & 31;
    int l0   = blockIdx.x * 128 + wave * 16;
    int col  = l0 + (lane & 15);
    int rowSel = (lane >> 4) * 2;                     // K placement: lanes 0-15 -> K0,K1 ; 16-31 -> K2,K3
    const float* xb = x + (size_t)b * C_ * L_ + col;

    v8f acc = {};                                     // D accumulator (16x16 f32)
    float sq = 0.f;
    for (int c0 = 0; c0 < C_; c0 += 4) {
        // B fragment: B[k][n] = x[b, c0+k, l0+n]
        float b0 = xb[(size_t)(c0 + rowSel) * L_];
        float b1 = xb[(size_t)(c0 + rowSel + 1) * L_];
        // A fragment: every row m holds g[c0..c0+3] -> all rows of D equal dot
        v2f Af; Af.x = sg[c0 + rowSel]; Af.y = sg[c0 + rowSel + 1];
        v2f Bf; Bf.x = b0;              Bf.y = b1;
        acc = __builtin_amdgcn_wmma_f32_16x16x4_f32(
            /*neg_a=*/false, Af, /*neg_b=*/false, Bf,
            /*c_mod=*/(short)0, acc, /*reuse_a=*/false, /*reuse_b=*/false);
        // per-lane partial sum of squares (half-wave split over K)
        sq = fmaf(b0, b0, sq);
        sq = fmaf(b1, b1, sq);
    }
    sq += __shfl_xor(sq, 16, 32);                     // fold the two K-halves
    float dot  = acc[0];                              // all D rows identical
    float xn   = sqrtf(sq);
    float cosv = dot / fmaxf(xn * gn[b], EPSV);
    if (lane < 16) cosb[b * L_ + col] = cosv;

    unsigned key = enc_key(cosv);
    unsigned kmin = key, kmax = key;
    #pragma unroll
    for (int off = 16; off > 0; off >>= 1) {
        unsigned a = __shfl_xor(kmin, off, 32);
        unsigned c = __shfl_xor(kmax, off, 32);
        kmin = (a < kmin) ? a : kmin;
        kmax = (c > kmax) ? c : kmax;
    }
    if (lane == 0) { atomicMin(&keys[0], kmin); atomicMax(&keys[1], kmax); }
}

// ---- K4: soft triangular histogram per batch, normalize, broadcast ----
__global__ void k_hist(const float* __restrict__ cosb, const unsigned* __restrict__ keys,
                       float* __restrict__ out) {
    int b = blockIdx.x, t = threadIdx.x;
    float minS = dec_key(keys[0]);
    float maxS = dec_key(keys[1]);
    float step = (maxS - minS) * (1.0f / (float)(NB - 1));

    float bins[NB];
    #pragma unroll
    for (int n = 0; n < NB; ++n) bins[n] = 0.f;
    for (int l = t; l < L_; l += 256) {
        float s = cosb[b * L_ + l];
        #pragma unroll
        for (int n = 0; n < NB; ++n) {
            float d = fabsf(s - (minS + (float)n * step));
            if (d < 0.03125f) bins[n] += (1.0f - d);   // 0.5/NB window, weight 1-d
        }
    }
    __shared__ float sb[NB];
    if (t < NB) sb[t] = 0.f;
    __syncthreads();
    #pragma unroll
    for (int n = 0; n < NB; ++n) atomicAdd(&sb[n], bins[n]);
    __syncthreads();
    float tot = 0.f;
    #pragma unroll
    for (int n = 0; n < NB; ++n) tot += sb[n];
    float inv = 1.0f / tot;
    // out[b, n, k] = count_map[b, n]; NB*(C_/NB) = 512 values per batch
    for (int i = t; i < NB * (C_ / NB); i += 256)
        out[b * (NB * (C_ / NB)) + i] = sb[i >> 5] * inv;   // C_/NB == 32
}

extern "C" void kernel_launch(void* const* d_in, const int* in_sizes, int n_in,
                              void* d_out, int out_size, void* d_ws, size_t ws_size,
                              hipStream_t stream) {
    const float* x = (const float*)d_in[0];
    float* out = (float*)d_out;
    char* ws = (char*)d_ws;
    float*    g    = (float*)(ws + WS_G);
    float*    gn   = (float*)(ws + WS_GN);
    unsigned* keys = (unsigned*)(ws + WS_KEYS);
    float*    cosb = (float*)(ws + WS_COS);

    k_init<<<1, 32, 0, stream>>>(keys);
    k_mean<<<(B_ * C_) / 8, 256, 0, stream>>>(x, g);
    k_gn<<<B_, 256, 0, stream>>>(g, gn);
    k_cos<<<dim3(L_ / 128, B_), 256, 0, stream>>>(x, g, gn, cosb, keys);
    k_hist<<<B_, 256, 0, stream>>>(cosb, keys, out);
}